// MPCPlanner_2156073583034
// MI455X (gfx1250) — compile-verified
//
#include <hip/hip_runtime.h>
#include <hip/hip_bf16.h>
#include <stdint.h>

// ---------------- problem constants ----------------
#define T_H   12
#define ITERS 10
#define CAND  1000
#define TOPK  100
#define A_DIM 6
#define B_DIM 16
#define H_DIM 200
#define Z_DIM 30
#define M_ROWS (B_DIM * CAND)          // 16000

// padded GEMM dims
#define XW    240                      // K for GEMM1: 200(b)+30(s)+6(a)+4 pad
#define HPAD  208                      // N for GEMM1 (13 tiles of 16)
#define NT1   13
#define KT1   (XW / 4)                 // 60
#define ZPAD  32                       // N for GEMM2 (2 tiles of 16)
#define NT2   2
#define KT2   (H_DIM / 4)              // 50

#define W1F_N (KT1 * NT1 * 64)         // 49920 floats, fragment-ordered
#define W2F_N (KT2 * NT2 * 64)         // 6400 floats, fragment-ordered
#define WAVES 6                        // waves per rollout block
#define NTILE ((M_ROWS + 15) / 16)     // 1000 row tiles

// ---------------- workspace layout (floats) ----------------
#define OFF_W1F   0
#define OFF_W2F   (OFF_W1F + W1F_N)
#define OFF_BB    (OFF_W2F + W2F_N)                  // 208
#define OFF_BZ    (OFF_BB + HPAD)                    // 32
#define OFF_WR    (OFF_BZ + ZPAD)                    // 232 (230 used)
#define OFF_MEAN  (OFF_WR + 232)                     // T*B*A = 1152
#define OFF_STD   (OFF_MEAN + T_H * B_DIM * A_DIM)   // 1152
#define OFF_RET   (OFF_STD + T_H * B_DIM * A_DIM)    // 16000
#define OFF_ACT   (OFF_RET + M_ROWS)                 // T*M*A = 1152000

typedef float v2f __attribute__((ext_vector_type(2)));
typedef float v8f __attribute__((ext_vector_type(8)));

__device__ __forceinline__ float fast_tanh(float x) {
    float cx = fminf(fmaxf(x, -12.0f), 12.0f);
    float e  = __expf(2.0f * cx);                     // v_exp_f32 path
    return (e - 1.0f) * __builtin_amdgcn_rcpf(e + 1.0f);
}

// ---------------- threefry2x32 (JAX-style key schedule) ----------------
__device__ __forceinline__ uint32_t rotl32(uint32_t x, int r) {
    return (x << r) | (x >> (32 - r));
}

__device__ void threefry2x32(uint32_t k0, uint32_t k1,
                             uint32_t c0, uint32_t c1,
                             uint32_t* o0, uint32_t* o1) {
    uint32_t ks0 = k0, ks1 = k1, ks2 = k0 ^ k1 ^ 0x1BD11BDAu;
    const int R0[4] = {13, 15, 26, 6};
    const int R1[4] = {17, 29, 16, 24};
    uint32_t x0 = c0 + ks0, x1 = c1 + ks1;
#define TF_ROUND4(R)                                    \
    {                                                   \
        for (int j = 0; j < 4; ++j) {                   \
            x0 += x1; x1 = rotl32(x1, R[j]); x1 ^= x0;  \
        }                                               \
    }
    TF_ROUND4(R0); x0 += ks1; x1 += ks2 + 1u;
    TF_ROUND4(R1); x0 += ks2; x1 += ks0 + 2u;
    TF_ROUND4(R0); x0 += ks0; x1 += ks1 + 3u;
    TF_ROUND4(R1); x0 += ks1; x1 += ks2 + 4u;
    TF_ROUND4(R0); x0 += ks2; x1 += ks0 + 5u;
#undef TF_ROUND4
    *o0 = x0; *o1 = x1;
}

// logical packed weight values
__device__ __forceinline__ float w1_val(int k, int n,
                                        const float* Wb, const float* Ws,
                                        const float* Wa) {
    if (n >= H_DIM) return 0.f;
    if (k < H_DIM)                 return Wb[k * H_DIM + n];
    if (k < H_DIM + Z_DIM)         return Ws[(k - H_DIM) * H_DIM + n];
    if (k < H_DIM + Z_DIM + A_DIM) return Wa[(k - H_DIM - Z_DIM) * H_DIM + n];
    return 0.f;
}

// ---------------- kernel 0: pack weights (fragment order) ----------------
__global__ void prep_kernel(const float* __restrict__ Wb,
                            const float* __restrict__ Ws,
                            const float* __restrict__ Wa,
                            const float* __restrict__ bb,
                            const float* __restrict__ Wz,
                            const float* __restrict__ bz,
                            const float* __restrict__ Wr,
                            float* __restrict__ ws) {
    int i = blockIdx.x * blockDim.x + threadIdx.x;
    if (i < W1F_N) {
        // fragment order: [kt][nt][lane][j]; B[k][n] with
        // k = kt*4 + 2*(lane>>4) + j, n = nt*16 + (lane&15)
        int kt = i / (NT1 * 64);
        int rem = i % (NT1 * 64);
        int nt = rem / 64;
        int q  = rem % 64;
        int l  = q >> 1, j = q & 1;
        int k = kt * 4 + ((l >> 4) << 1) + j;
        int n = nt * 16 + (l & 15);
        ws[OFF_W1F + i] = w1_val(k, n, Wb, Ws, Wa);
        return;
    }
    i -= W1F_N;
    if (i < W2F_N) {
        int kt = i / (NT2 * 64);
        int rem = i % (NT2 * 64);
        int nt = rem / 64;
        int q  = rem % 64;
        int l  = q >> 1, j = q & 1;
        int k = kt * 4 + ((l >> 4) << 1) + j;
        int n = nt * 16 + (l & 15);
        ws[OFF_W2F + i] = (k < H_DIM && n < Z_DIM) ? Wz[k * Z_DIM + n] : 0.f;
        return;
    }
    i -= W2F_N;
    if (i < HPAD) { ws[OFF_BB + i] = (i < H_DIM) ? bb[i] : 0.f; return; }
    i -= HPAD;
    if (i < ZPAD) { ws[OFF_BZ + i] = (i < Z_DIM) ? bz[i] : 0.f; return; }
    i -= ZPAD;
    if (i < 232)  { ws[OFF_WR + i] = (i < H_DIM + Z_DIM) ? Wr[i] : 0.f; return; }
    i -= 232;
    if (i < T_H * B_DIM * A_DIM) { ws[OFF_MEAN + i] = 0.f; return; }
    i -= T_H * B_DIM * A_DIM;
    if (i < T_H * B_DIM * A_DIM) { ws[OFF_STD + i] = 1.f; return; }
}

// ---------------- kernel 1: sample actions ----------------
__global__ void gen_actions_kernel(const float* __restrict__ mean,
                                   const float* __restrict__ stdv,
                                   float* __restrict__ actions, int iter) {
    int i = blockIdx.x * blockDim.x + threadIdx.x;
    const int total = T_H * M_ROWS * A_DIM;
    if (i >= total) return;
    int a = i % A_DIM;
    int m = (i / A_DIM) % M_ROWS;
    int t = i / (A_DIM * M_ROWS);
    int b = m / CAND;
    uint32_t k0, k1, r0, r1;
    threefry2x32(0u, 42u, 0u, (uint32_t)iter, &k0, &k1);       // fold_in
    threefry2x32(k0, k1, (uint32_t)i, 0x9E3779B9u, &r0, &r1);
    float u1 = fmaxf((r0 >> 8) * (1.0f / 16777216.0f), 1.1920929e-7f);
    float u2 = (r1 >> 8) * (1.0f / 16777216.0f);
    float n  = sqrtf(-2.0f * __logf(u1)) * __cosf(6.28318530718f * u2);
    int mi = (t * B_DIM + b) * A_DIM + a;
    actions[i] = mean[mi] + stdv[mi] * n;
}

// ---------------- kernel 2: WMMA rollout + returns ----------------
// 192 threads (6 waves); each wave owns one 16-row tile; all weights in LDS.
__global__ __launch_bounds__(192)
void rollout_kernel(const float* __restrict__ belief,
                    const float* __restrict__ state,
                    const float* __restrict__ W1F,
                    const float* __restrict__ W2F,
                    const float* __restrict__ bbP,
                    const float* __restrict__ bzP,
                    const float* __restrict__ WrP,
                    const float* __restrict__ br,
                    const float* __restrict__ actions,
                    float* __restrict__ returns) {
    extern __shared__ float smem[];
    float* W1s = smem;                         // 49920 floats (199.7 KB)
    float* W2s = smem + W1F_N;                 //  6400 floats ( 25.6 KB)
    float* Xall = smem + W1F_N + W2F_N;        // 6*16*240     ( 92.2 KB)

    const int tid  = threadIdx.x;
    const int lane = tid & 31;
    const int wave = tid >> 5;

    // ---- cooperative coalesced load of fragment-ordered weights ----
    for (int i = tid; i < W1F_N; i += 192) W1s[i] = W1F[i];
    for (int i = tid; i < W2F_N; i += 192) W2s[i] = W2F[i];
    __syncthreads();

    const int tile = blockIdx.x * WAVES + wave;
    if (tile >= NTILE) return;
    const int row0 = tile * 16;
    float* Xw = Xall + wave * (16 * XW);

    const int rr   = lane & 15;
    const int half = lane >> 4;
    const float brv = br[0];

    // ---- init X = [belief | state | 0 | 0] ----
    for (int idx = lane; idx < 16 * XW; idx += 32) {
        int r = idx / XW, c = idx % XW;
        int bidx = (row0 + r) / CAND;
        float v = 0.f;
        if (c < H_DIM)              v = belief[bidx * H_DIM + c];
        else if (c < H_DIM + Z_DIM) v = state[bidx * Z_DIM + (c - H_DIM)];
        Xw[r * XW + c] = v;
    }
    asm volatile("s_wait_dscnt 0" ::: "memory");

    float retAcc = 0.f;

    for (int t = 0; t < T_H; ++t) {
        // ---- load a_t into X cols [230,236) ----
        const float* ap = actions + ((size_t)t * M_ROWS + row0) * A_DIM;
        for (int idx = lane; idx < 16 * A_DIM; idx += 32) {
            int r = idx / A_DIM, c = idx % A_DIM;
            Xw[r * XW + H_DIM + Z_DIM + c] = ap[r * A_DIM + c];
        }
        asm volatile("s_wait_dscnt 0" ::: "memory");

        // ---- GEMM1: [16 x 240] @ [240 x 208], all operands from LDS ----
        v8f acc[NT1];
#pragma unroll
        for (int i = 0; i < NT1; ++i)
#pragma unroll
            for (int e = 0; e < 8; ++e) acc[i][e] = 0.f;

        for (int kt = 0; kt < KT1; ++kt) {
            int kk = kt * 4 + (half << 1);
            v2f aF = *(const v2f*)&Xw[rr * XW + kk];           // ds_load_b64
            const float* wrow = &W1s[(kt * NT1) * 64 + (lane << 1)];
            // explicit unroll so the RA reuse hint is a literal constant
#define G1_STEP(NT, RA)                                               \
            {                                                         \
                v2f bF = *(const v2f*)&wrow[(NT) * 64];               \
                acc[NT] = __builtin_amdgcn_wmma_f32_16x16x4_f32(      \
                    false, aF, false, bF, (short)0, acc[NT], RA, false); \
            }
            G1_STEP(0, false)
            G1_STEP(1, true)  G1_STEP(2, true)  G1_STEP(3, true)
            G1_STEP(4, true)  G1_STEP(5, true)  G1_STEP(6, true)
            G1_STEP(7, true)  G1_STEP(8, true)  G1_STEP(9, true)
            G1_STEP(10, true) G1_STEP(11, true) G1_STEP(12, true)
#undef G1_STEP
        }

        // ---- bias + tanh, write b_new back into X cols [0,200) ----
#pragma unroll
        for (int nt = 0; nt < NT1; ++nt) {
            int n = nt * 16 + rr;
            float bias = bbP[n];
#pragma unroll
            for (int e = 0; e < 8; ++e) {
                int m = e + (half << 3);
                float v = fast_tanh(acc[nt][e] + bias);
                if (n < H_DIM) Xw[m * XW + n] = v;
            }
        }
        asm volatile("s_wait_dscnt 0" ::: "memory");

        // ---- GEMM2: s_new = tanh(b_new[16x200] @ Wz[200x32] + bz) ----
        v8f acc2[NT2];
#pragma unroll
        for (int i = 0; i < NT2; ++i)
#pragma unroll
            for (int e = 0; e < 8; ++e) acc2[i][e] = 0.f;

        for (int kt = 0; kt < KT2; ++kt) {
            int kk = kt * 4 + (half << 1);
            v2f aF = *(const v2f*)&Xw[rr * XW + kk];
            const float* wrow = &W2s[(kt * NT2) * 64 + (lane << 1)];
            {
                v2f bF = *(const v2f*)&wrow[0];
                acc2[0] = __builtin_amdgcn_wmma_f32_16x16x4_f32(
                    false, aF, false, bF, (short)0, acc2[0], false, false);
            }
            {
                v2f bF = *(const v2f*)&wrow[64];
                acc2[1] = __builtin_amdgcn_wmma_f32_16x16x4_f32(
                    false, aF, false, bF, (short)0, acc2[1], true, false);
            }
        }
#pragma unroll
        for (int nt = 0; nt < NT2; ++nt) {
            int n = nt * 16 + rr;
            float bias = bzP[n];
#pragma unroll
            for (int e = 0; e < 8; ++e) {
                int m = e + (half << 3);
                float v = fast_tanh(acc2[nt][e] + bias);
                if (n < Z_DIM) Xw[m * XW + H_DIM + n] = v;
            }
        }
        asm volatile("s_wait_dscnt 0" ::: "memory");

        // ---- reward: ret += [b_new|s_new] . Wr + br ----
        {
            int c0 = half * 115;
            float p = 0.f;
            for (int c = 0; c < 115; ++c)
                p += Xw[rr * XW + c0 + c] * WrP[c0 + c];
            retAcc += p;
            if (half == 0) retAcc += brv;
        }
    }

    retAcc += __shfl_xor(retAcc, 16, 32);
    if (lane < 16) returns[row0 + lane] = retAcc;
}

// ---------------- kernel 3: top-k + mean/std refit ----------------
__global__ __launch_bounds__(128)
void topk_kernel(const float* __restrict__ returns,
                 const float* __restrict__ actions,
                 float* __restrict__ mean,
                 float* __restrict__ stdv) {
    __shared__ float vals[CAND];
    __shared__ int   selIdx[TOPK];
    __shared__ float redV[128];
    __shared__ int   redI[128];

    const int b   = blockIdx.x;
    const int tid = threadIdx.x;

    for (int i = tid; i < CAND; i += 128) vals[i] = returns[b * CAND + i];
    __syncthreads();

    for (int k = 0; k < TOPK; ++k) {
        float mv = -3.4e38f; int mi = -1;
        for (int i = tid; i < CAND; i += 128)
            if (vals[i] > mv) { mv = vals[i]; mi = i; }
        redV[tid] = mv; redI[tid] = mi;
        __syncthreads();
        for (int s = 64; s > 0; s >>= 1) {
            if (tid < s && redV[tid + s] > redV[tid]) {
                redV[tid] = redV[tid + s]; redI[tid] = redI[tid + s];
            }
            __syncthreads();
        }
        if (tid == 0) { selIdx[k] = redI[0]; vals[redI[0]] = -3.4e38f; }
        __syncthreads();
    }

    for (int pair = tid; pair < T_H * A_DIM; pair += 128) {
        int t = pair / A_DIM, a = pair % A_DIM;
        float sum = 0.f, ssq = 0.f;
        for (int j = 0; j < TOPK; ++j) {
            int idx = selIdx[j];
            float v = actions[((size_t)t * M_ROWS + b * CAND + idx) * A_DIM + a];
            sum += v; ssq += v * v;
        }
        float m = sum * (1.0f / TOPK);
        float var = ssq * (1.0f / TOPK) - m * m;
        int mi = (t * B_DIM + b) * A_DIM + a;
        mean[mi] = m;
        stdv[mi] = sqrtf(fmaxf(var, 0.f));
    }
}

// ---------------- kernel 4: emit mean[t=0] ----------------
__global__ void finalize_kernel(const float* __restrict__ mean,
                                float* __restrict__ out) {
    int i = threadIdx.x;
    if (i < B_DIM * A_DIM) out[i] = mean[i];
}

// ---------------- host launcher ----------------
extern "C" void kernel_launch(void* const* d_in, const int* in_sizes, int n_in,
                              void* d_out, int out_size, void* d_ws, size_t ws_size,
                              hipStream_t stream) {
    const float* belief = (const float*)d_in[0];
    const float* state  = (const float*)d_in[1];
    const float* Wb     = (const float*)d_in[2];
    const float* Ws     = (const float*)d_in[3];
    const float* Wa     = (const float*)d_in[4];
    const float* bb     = (const float*)d_in[5];
    const float* Wz     = (const float*)d_in[6];
    const float* bz     = (const float*)d_in[7];
    const float* Wr     = (const float*)d_in[8];
    const float* br     = (const float*)d_in[9];

    float* ws = (float*)d_ws;
    float* W1F  = ws + OFF_W1F;
    float* W2F  = ws + OFF_W2F;
    float* bbP  = ws + OFF_BB;
    float* bzP  = ws + OFF_BZ;
    float* WrP  = ws + OFF_WR;
    float* mean = ws + OFF_MEAN;
    float* stdv = ws + OFF_STD;
    float* ret  = ws + OFF_RET;
    float* act  = ws + OFF_ACT;

    const int prepN = W1F_N + W2F_N + HPAD + ZPAD + 232
                    + 2 * T_H * B_DIM * A_DIM;
    prep_kernel<<<(prepN + 255) / 256, 256, 0, stream>>>(Wb, Ws, Wa, bb, Wz, bz,
                                                          Wr, ws);

    const int genN = T_H * M_ROWS * A_DIM;
    const int rollBlocks = (NTILE + WAVES - 1) / WAVES;               // 167
    const size_t smemBytes =
        (size_t)(W1F_N + W2F_N + WAVES * 16 * XW) * sizeof(float);    // 317440

    for (int it = 0; it < ITERS; ++it) {
        gen_actions_kernel<<<(genN + 255) / 256, 256, 0, stream>>>(mean, stdv,
                                                                    act, it);
        rollout_kernel<<<rollBlocks, 192, smemBytes, stream>>>(
            belief, state, W1F, W2F, bbP, bzP, WrP, br, act, ret);
        topk_kernel<<<B_DIM, 128, 0, stream>>>(ret, act, mean, stdv);
    }
    finalize_kernel<<<1, 96, 0, stream>>>(mean, (float*)d_out);
}